// CEtransitionLoss_15479062135236
// MI455X (gfx1250) — compile-verified
//
#include <hip/hip_runtime.h>
#include <stdint.h>

#define B_DIM 512
#define C_DIM 3
#define T_DIM 16384
#define CHUNK_T 1024
#define TOTAL_CHUNKS (B_DIM * (T_DIM / CHUNK_T))   // 8192
#define BLOCK_THREADS 256
#define NBLOCKS 4096
#define SMOOTH_W 0.01
#define TRANS_W 0.1
#define USE_ASYNC_LDS 1

// forbidden transitions (prev,next) in {(0,2),(2,1),(1,0)} -> next == prev+2 mod 3
// bitmask over idx = prev*3+next : bits {2,7,3} set
#define FORBID_MASK 0x8Cu

__device__ __forceinline__ int argmax3(float a, float b, float c) {
    int p = 0; float m = a;
    if (b > m) { m = b; p = 1; }
    if (c > m) { p = 2; }
    return p;
}

__device__ __forceinline__ float waveSum(float v) {
#pragma unroll
    for (int o = 16; o > 0; o >>= 1) v += __shfl_down(v, o, 32);
    return v;
}

__global__ __launch_bounds__(BLOCK_THREADS)
void ce_trans_loss_pass1(const float* __restrict__ logits,
                         const int*   __restrict__ labels,
                         float*       __restrict__ partials,
                         int nblocks)
{
    __shared__ __align__(16) float tile[C_DIM][CHUNK_T];
    __shared__ float redCe[8], redSm[8], redTr[8];

    const int tid = threadIdx.x;
    float ceAcc = 0.0f, smAcc = 0.0f;
    int trAcc = 0;

    for (int chunk = blockIdx.x; chunk < TOTAL_CHUNKS; chunk += nblocks) {
        const int b   = chunk >> 4;          // chunk / 16
        const int t0c = (chunk & 15) << 10;  // (chunk % 16) * 1024
        const int t0  = t0c + tid * 4;
        const size_t rowBase = (size_t)b * C_DIM * T_DIM;

        // ---- stage 3 x 1024 logits into LDS ----
#if USE_ASYNC_LDS
#pragma unroll
        for (int c = 0; c < C_DIM; ++c) {
            const float* gsrc = logits + rowBase + (size_t)c * T_DIM + t0;
            unsigned ldsOff = (unsigned)(uintptr_t)(&tile[c][tid * 4]);
            asm volatile("global_load_async_to_lds_b128 %0, %1, off"
                         :: "v"(ldsOff), "v"(gsrc) : "memory");
        }
        asm volatile("s_wait_asynccnt 0" ::: "memory");
#else
#pragma unroll
        for (int c = 0; c < C_DIM; ++c) {
            const float4 v = *(const float4*)(logits + rowBase + (size_t)c * T_DIM + t0);
            *(float4*)(&tile[c][tid * 4]) = v;
        }
#endif
        __syncthreads();

        const int4 lab4 = *(const int4*)(labels + (size_t)b * T_DIM + t0);

        const float4 x0 = *(const float4*)(&tile[0][tid * 4]);
        const float4 x1 = *(const float4*)(&tile[1][tid * 4]);
        const float4 x2 = *(const float4*)(&tile[2][tid * 4]);

        // neighbor at t0-1 (from LDS within chunk; lane 0 of chunk loads globally)
        const bool validPrev = (t0 > 0);
        float p0v = 0.f, p1v = 0.f, p2v = 0.f;
        if (tid > 0) {
            p0v = tile[0][tid * 4 - 1];
            p1v = tile[1][tid * 4 - 1];
            p2v = tile[2][tid * 4 - 1];
        } else if (validPrev) {
            p0v = logits[rowBase + 0 * T_DIM + t0 - 1];
            p1v = logits[rowBase + 1 * T_DIM + t0 - 1];
            p2v = logits[rowBase + 2 * T_DIM + t0 - 1];
        }

        const float a[4]  = {x0.x, x0.y, x0.z, x0.w};
        const float bb[4] = {x1.x, x1.y, x1.z, x1.w};
        const float cc[4] = {x2.x, x2.y, x2.z, x2.w};
        const int   lb[4] = {lab4.x, lab4.y, lab4.z, lab4.w};

        int pr[4];
#pragma unroll
        for (int j = 0; j < 4; ++j) {
            const float va = a[j], vb = bb[j], vc = cc[j];
            const float m   = fmaxf(va, fmaxf(vb, vc));
            const float s   = __expf(va - m) + __expf(vb - m) + __expf(vc - m);
            const float lse = m + __logf(s);
            const float picked = (lb[j] == 0) ? va : ((lb[j] == 1) ? vb : vc);
            ceAcc += lse - picked;
            pr[j] = argmax3(va, vb, vc);
        }

        if (validPrev) {
            const float d0 = a[0] - p0v, d1 = bb[0] - p1v, d2 = cc[0] - p2v;
            smAcc += d0 * d0 + d1 * d1 + d2 * d2;
            const int pp = argmax3(p0v, p1v, p2v);
            trAcc += (int)((FORBID_MASK >> (pp * 3 + pr[0])) & 1u);
        }
#pragma unroll
        for (int j = 1; j < 4; ++j) {
            const float d0 = a[j] - a[j - 1], d1 = bb[j] - bb[j - 1], d2 = cc[j] - cc[j - 1];
            smAcc += d0 * d0 + d1 * d1 + d2 * d2;
            trAcc += (int)((FORBID_MASK >> (pr[j - 1] * 3 + pr[j])) & 1u);
        }

        __syncthreads();   // tile fully consumed; safe to restage next chunk
    }

    // ---- deterministic block reduction (wave32 shuffles + LDS) ----
    float ce = waveSum(ceAcc);
    float sm = waveSum(smAcc);
    float tr = waveSum((float)trAcc);
    const int lane = tid & 31, wid = tid >> 5;
    if (lane == 0) { redCe[wid] = ce; redSm[wid] = sm; redTr[wid] = tr; }
    __syncthreads();
    if (wid == 0) {
        float c2 = (lane < 8) ? redCe[lane] : 0.f;
        float s2 = (lane < 8) ? redSm[lane] : 0.f;
        float t2 = (lane < 8) ? redTr[lane] : 0.f;
#pragma unroll
        for (int o = 4; o > 0; o >>= 1) {
            c2 += __shfl_down(c2, o, 32);
            s2 += __shfl_down(s2, o, 32);
            t2 += __shfl_down(t2, o, 32);
        }
        if (lane == 0) {
            partials[blockIdx.x]                = c2;
            partials[nblocks + blockIdx.x]      = s2;
            partials[2 * nblocks + blockIdx.x]  = t2;
        }
    }
}

__global__ __launch_bounds__(256)
void ce_trans_loss_pass2(const float* __restrict__ partials,
                         float* __restrict__ out, int nblocks)
{
    __shared__ double sce[256], ssm[256], str[256];
    const int tid = threadIdx.x;
    double ce = 0.0, sm = 0.0, tr = 0.0;
    for (int i = tid; i < nblocks; i += 256) {
        ce += (double)partials[i];
        sm += (double)partials[nblocks + i];
        tr += (double)partials[2 * nblocks + i];
    }
    sce[tid] = ce; ssm[tid] = sm; str[tid] = tr;
    __syncthreads();
    for (int s = 128; s > 0; s >>= 1) {
        if (tid < s) {
            sce[tid] += sce[tid + s];
            ssm[tid] += ssm[tid + s];
            str[tid] += str[tid + s];
        }
        __syncthreads();
    }
    if (tid == 0) {
        const double ceMean = sce[0] / ((double)B_DIM * (double)T_DIM);
        const double smTerm = SMOOTH_W * ssm[0] /
                              ((double)B_DIM * (double)C_DIM * (double)(T_DIM - 1));
        const double trTerm = (str[0] > 0.5) ? TRANS_W : 0.0;
        out[0] = (float)(ceMean + smTerm + trTerm);
    }
}

extern "C" void kernel_launch(void* const* d_in, const int* in_sizes, int n_in,
                              void* d_out, int out_size, void* d_ws, size_t ws_size,
                              hipStream_t stream)
{
    (void)in_sizes; (void)n_in; (void)out_size; (void)ws_size;
    const float* logits = (const float*)d_in[0];
    const int*   labels = (const int*)d_in[1];
    float* partials = (float*)d_ws;          // needs 3 * NBLOCKS * 4 = 48 KB
    float* out      = (float*)d_out;

    ce_trans_loss_pass1<<<NBLOCKS, BLOCK_THREADS, 0, stream>>>(logits, labels, partials, NBLOCKS);
    ce_trans_loss_pass2<<<1, 256, 0, stream>>>(partials, out, NBLOCKS);
}